// S1AttInform_67568425501182
// MI455X (gfx1250) — compile-verified
//
#include <hip/hip_runtime.h>

typedef float v2f __attribute__((ext_vector_type(2)));
typedef float v8f __attribute__((ext_vector_type(8)));

__device__ __forceinline__ v8f wmma4(v2f a, v2f b, v8f c) {
  // V_WMMA_F32_16X16X4_F32: D = A(16x4) x B(4x16) + C(16x16), full fp32
  return __builtin_amdgcn_wmma_f32_16x16x4_f32(false, a, false, b, (short)0, c,
                                               false, false);
}

__device__ __forceinline__ float lrelu(float v) { return v > 0.f ? v : 0.01f * v; }

// ---------------------------------------------------------------------------
// Conv2d (5,1) stride (2,1) pad (2,0) + bias. x:[N,32,40,7] -> h0:[N,16,20,7]
// ---------------------------------------------------------------------------
__global__ void conv_time(const float* __restrict__ x, const float* __restrict__ w,
                          const float* __restrict__ bias, float* __restrict__ h0) {
  int idx = blockIdx.x * blockDim.x + threadIdx.x;
  const int TOT = 1024 * 16 * 20 * 7;
  if (idx >= TOT) return;
  int v = idx % 7;
  int t = (idx / 7) % 20;
  int c = (idx / 140) % 16;
  int n = idx / 2240;
  float acc = bias[c];
  for (int d = 0; d < 32; ++d) {
    const float* xp = x + (((size_t)n * 32 + d) * 40) * 7 + v;
    const float* wp = w + ((size_t)c * 32 + d) * 5;
#pragma unroll
    for (int k = 0; k < 5; ++k) {
      int tt = 2 * t + k - 2;
      if (tt >= 0 && tt < 40) acc += xp[tt * 7] * wp[k];
    }
  }
  h0[idx] = acc;  // layout [n][c][t][v]
}

// ---------------------------------------------------------------------------
// BN2d batch stats per channel (biased var) -> scale/shift (16 channels)
// ---------------------------------------------------------------------------
__global__ void bn2d_stats(const float* __restrict__ h0, const float* __restrict__ g,
                           const float* __restrict__ b, float* __restrict__ scale,
                           float* __restrict__ shift) {
  __shared__ float ss[256], sq[256];
  int c = blockIdx.x;   // 16 blocks
  int tid = threadIdx.x;  // 256 threads
  const int PER = 140;    // 20*7
  float s = 0.f, q = 0.f;
  for (int idx = tid; idx < 1024 * PER; idx += 256) {
    int n = idx / PER, i = idx - n * PER;
    float v = h0[((size_t)n * 16 + c) * PER + i];
    s += v;
    q += v * v;
  }
  ss[tid] = s; sq[tid] = q;
  __syncthreads();
  for (int w = 128; w > 0; w >>= 1) {
    if (tid < w) { ss[tid] += ss[tid + w]; sq[tid] += sq[tid + w]; }
    __syncthreads();
  }
  if (tid == 0) {
    float cnt = 1024.f * PER;
    float mean = ss[0] / cnt;
    float var = sq[0] / cnt - mean * mean;
    float inv = rsqrtf(var + 1e-5f);
    float sc = inv * g[c];
    scale[c] = sc;
    shift[c] = b[c] - mean * sc;
  }
}

// apply BN2d and transpose NCHW -> [n][t][v][c] (so x1/x2 are plain views)
__global__ void bn2d_apply_nhwc(const float* __restrict__ h0,
                                const float* __restrict__ scale,
                                const float* __restrict__ shift,
                                float* __restrict__ h) {
  int idx = blockIdx.x * blockDim.x + threadIdx.x;
  const int TOT = 1024 * 2240;
  if (idx >= TOT) return;
  int c = idx & 15;
  int v = (idx >> 4) % 7;
  int t = (idx / 112) % 20;
  int n = idx / 2240;
  float val = h0[((size_t)(n * 16 + c) * 20 + t) * 7 + v];
  h[idx] = val * scale[c] + shift[c];
}

// ---------------------------------------------------------------------------
// Generic WMMA fp32 GEMM, M-blocked x2: each block computes a 32-row tile;
// each wave holds two 16x16 accumulators sharing one B fragment per k-step.
// C[R,O] (+)= (opt lrelu)(A[R,K] @ W[K,O] + bias)
// flags: 1 = accumulate into existing C, 2 = leaky_relu on output
// Requirements: R % 32 == 0, lda even. OOB columns clamped to col 0
// (their lanes' accumulators hold garbage that is never stored).
// ---------------------------------------------------------------------------
__global__ void gemm_wmma_f32(const float* __restrict__ A, int lda,
                              const float* __restrict__ W, int ldw,
                              const float* __restrict__ bias, float* __restrict__ C,
                              int ldc, int R, int K, int O, int flags) {
  int rt = blockIdx.x;  // 32-row tile
  int ct = blockIdx.y * 4 + threadIdx.y;
  if (ct * 16 >= O) return;  // uniform per wave
  int lane = threadIdx.x;
  int half = lane >> 4;
  int lrow = lane & 15;
  int row0 = rt * 32 + lrow;
  int col = ct * 16 + lrow;
  bool colok = col < O;
  int colc = colok ? col : 0;  // clamp: OOB lanes read col 0 (harmless)

  v8f acc0, acc1;
  if (flags & 1) {
#pragma unroll
    for (int i = 0; i < 8; ++i) {
      int rr = rt * 32 + i + 8 * half;
      acc0[i] = C[(size_t)rr * ldc + colc];
      acc1[i] = C[(size_t)(rr + 16) * ldc + colc];
    }
  } else {
    float b0 = (bias != nullptr) ? bias[colc] : 0.f;
#pragma unroll
    for (int i = 0; i < 8; ++i) { acc0[i] = b0; acc1[i] = b0; }
  }

  const float* A0 = A + (size_t)row0 * lda + 2 * half;
  const float* A1 = A0 + (size_t)16 * lda;
  const float* Wc = W + colc + (size_t)(2 * half) * ldw;
  int nk4 = K >> 2;  // guard-free main loop
  for (int ks = 0; ks < nk4; ++ks) {
    v2f a0 = *(const v2f*)(A0 + 4 * ks);
    v2f a1 = *(const v2f*)(A1 + 4 * ks);
    v2f b;
    b.x = Wc[(size_t)(4 * ks) * ldw];
    b.y = Wc[(size_t)(4 * ks + 1) * ldw];
    acc0 = wmma4(a0, b, acc0);
    acc1 = wmma4(a1, b, acc1);
  }
  if (K & 3) {  // peeled tail (only K=126 in this net): clamped idx + masks
    int kb = K & ~3;
    int ka = kb + 2 * half;
    float m0 = (ka < K) ? 1.f : 0.f;
    float m1 = (ka + 1 < K) ? 1.f : 0.f;
    int k0 = (ka < K) ? ka : 0;
    int k1 = (ka + 1 < K) ? ka + 1 : 0;
    const float* Ar0 = A + (size_t)row0 * lda;
    const float* Ar1 = Ar0 + (size_t)16 * lda;
    const float* Wb = W + colc;
    v2f a0, a1, b;
    a0.x = Ar0[k0] * m0;
    a0.y = Ar0[k1] * m1;
    a1.x = Ar1[k0] * m0;
    a1.y = Ar1[k1] * m1;
    b.x = Wb[(size_t)k0 * ldw] * m0;
    b.y = Wb[(size_t)k1 * ldw] * m1;
    acc0 = wmma4(a0, b, acc0);
    acc1 = wmma4(a1, b, acc1);
  }

  bool act = (flags & 2) != 0;
#pragma unroll
  for (int i = 0; i < 8; ++i) {
    int rr = rt * 32 + i + 8 * half;
    if (colok) {
      float v0 = acc0[i];
      float v1 = acc1[i];
      if (act) { v0 = lrelu(v0); v1 = lrelu(v1); }
      C[(size_t)rr * ldc + col] = v0;
      C[(size_t)(rr + 16) * ldc + col] = v1;
    }
  }
}

// ---------------------------------------------------------------------------
// Fused edge MLP, M-blocked x2 (32 edge rows / block):
// gather [r, r-s] into LDS, fc1 (WMMA+lrelu) to LDS,
// fc2 = Y1@W2[0:H] + A@W2[H:H+2C] (+bias) -> Y2 (pre-BN) to global.
// X:[Nn,Vn,C]  W1:[2C,H]  W2:[H+2C,H]  Y2:[Nn*E,H]   (E = Vn*(Vn-1), O==H)
// F=2C and Hp are multiples of 4; loops are guard-free (cols clamped).
// ---------------------------------------------------------------------------
__global__ void edge_mlp_fused(const float* __restrict__ X, const float* __restrict__ W1,
                               const float* __restrict__ b1, const float* __restrict__ W2,
                               const float* __restrict__ b2, float* __restrict__ Y2,
                               int E, int Vn, int C, int H) {
  extern __shared__ float lds[];
  const int F = 2 * C;
  const int Hp = (H + 15) & ~15;
  float* Alds = lds;            // [32][F]
  float* Y1 = lds + 32 * F;     // [32][Hp]
  int rt = blockIdx.x;          // 32-row tile
  int tid = threadIdx.y * 32 + threadIdx.x;

  for (int idx = tid; idx < 32 * F; idx += 256) {
    int r = idx / F, k = idx - r * F;
    int grow = rt * 32 + r;
    int n = grow / E, e = grow - n * E;
    int i = e / (Vn - 1), jj = e - i * (Vn - 1);
    int j = jj + (jj >= i ? 1 : 0);
    const float* Xi = X + ((size_t)n * Vn + i) * C;
    float val;
    if (k < C) val = Xi[k];
    else {
      int kk = k - C;
      val = Xi[kk] - X[((size_t)n * Vn + j) * C + kk];
    }
    Alds[idx] = val;
  }
  __syncthreads();

  int lane = threadIdx.x, half = lane >> 4, lrow = lane & 15;
  int ct = threadIdx.y;  // 8 waves -> 8 col tiles (covers H<=128)
  int col = ct * 16 + lrow;
  bool colok = col < H;
  int colc = colok ? col : 0;  // clamp

  const float* AL0 = Alds + lrow * F + 2 * half;  // 8B aligned (F even)
  const float* AL1 = AL0 + 16 * F;
  const float* Y1L0 = Y1 + lrow * Hp + 2 * half;
  const float* Y1L1 = Y1L0 + 16 * Hp;

  // fc1
  v8f acc0, acc1;
  {
    float b0 = b1[colc];
#pragma unroll
    for (int i = 0; i < 8; ++i) { acc0[i] = b0; acc1[i] = b0; }
  }
  {
    const float* W1c = W1 + colc + (size_t)(2 * half) * H;
    for (int ks = 0; ks < F / 4; ++ks) {
      v2f a0 = *(const v2f*)(AL0 + 4 * ks);
      v2f a1 = *(const v2f*)(AL1 + 4 * ks);
      v2f b;
      b.x = W1c[(size_t)(4 * ks) * H];
      b.y = W1c[(size_t)(4 * ks + 1) * H];
      acc0 = wmma4(a0, b, acc0);
      acc1 = wmma4(a1, b, acc1);
    }
  }
#pragma unroll
  for (int i = 0; i < 8; ++i) {
    int rr = i + 8 * half;
    float v0 = lrelu(acc0[i]);
    float v1 = lrelu(acc1[i]);
    // Y1 feeds fc2 as the A operand: padding cols H..Hp-1 MUST be zero.
    Y1[rr * Hp + col] = colok ? v0 : 0.f;         // value select, no branch
    Y1[(rr + 16) * Hp + col] = colok ? v1 : 0.f;
  }
  __syncthreads();

  // fc2 = Y1 @ W2[0:H]  +  A @ W2[H:H+F]  + b2
  v8f acc2a, acc2b;
  {
    float b0 = b2[colc];
#pragma unroll
    for (int i = 0; i < 8; ++i) { acc2a[i] = b0; acc2b[i] = b0; }
  }
  {
    const float* W2c = W2 + colc + (size_t)(2 * half) * H;
    for (int ks = 0; ks < Hp / 4; ++ks) {
      v2f a0 = *(const v2f*)(Y1L0 + 4 * ks);
      v2f a1 = *(const v2f*)(Y1L1 + 4 * ks);
      v2f b;
      // rows ka>=H multiply the zeroed Y1 padding -> contribute 0 regardless
      b.x = W2c[(size_t)(4 * ks) * H];
      b.y = W2c[(size_t)(4 * ks + 1) * H];
      acc2a = wmma4(a0, b, acc2a);
      acc2b = wmma4(a1, b, acc2b);
    }
    const float* W2x = W2 + (size_t)H * H + colc + (size_t)(2 * half) * H;
    for (int ks = 0; ks < F / 4; ++ks) {
      v2f a0 = *(const v2f*)(AL0 + 4 * ks);
      v2f a1 = *(const v2f*)(AL1 + 4 * ks);
      v2f b;
      b.x = W2x[(size_t)(4 * ks) * H];
      b.y = W2x[(size_t)(4 * ks + 1) * H];
      acc2a = wmma4(a0, b, acc2a);
      acc2b = wmma4(a1, b, acc2b);
    }
  }
#pragma unroll
  for (int i = 0; i < 8; ++i) {
    int rr = rt * 32 + i + 8 * half;
    if (colok) {
      Y2[(size_t)rr * H + col] = acc2a[i];
      Y2[(size_t)(rr + 16) * H + col] = acc2b[i];
    }
  }
}

// ---------------------------------------------------------------------------
// BN1d batch stats over rows (deterministic two-stage column reduction)
// ---------------------------------------------------------------------------
__global__ void colsum_partial(const float* __restrict__ Y, int R, int O,
                               float* __restrict__ part) {
  int t = threadIdx.x;  // 128 threads = columns
  float s = 0.f, q = 0.f;
  if (t < O) {
    for (int r = blockIdx.x; r < R; r += gridDim.x) {
      float v = Y[(size_t)r * O + t];
      s += v;
      q += v * v;
    }
  }
  part[blockIdx.x * 256 + t] = s;
  part[blockIdx.x * 256 + 128 + t] = q;
}

__global__ void colstats_final(const float* __restrict__ part, int nparts, int R, int O,
                               const float* __restrict__ g, const float* __restrict__ b,
                               float* __restrict__ scale, float* __restrict__ shift) {
  int t = threadIdx.x;
  if (t >= O) return;
  float s = 0.f, q = 0.f;
  for (int p = 0; p < nparts; ++p) {
    s += part[p * 256 + t];
    q += part[p * 256 + 128 + t];
  }
  float mean = s / (float)R;
  float var = q / (float)R - mean * mean;  // biased var
  float inv = rsqrtf(var + 1e-5f);
  float sc = inv * g[t];
  scale[t] = sc;
  shift[t] = b[t] - mean * sc;
}

__global__ void bn_apply_lrelu(const float* __restrict__ Y, const float* __restrict__ scale,
                               const float* __restrict__ shift, float* __restrict__ out,
                               int O, int total) {
  int idx = blockIdx.x * blockDim.x + threadIdx.x;
  if (idx >= total) return;
  int c = idx % O;
  out[idx] = lrelu(Y[idx] * scale[c] + shift[c]);
}

// edge2node mean (recv groups are contiguous) fused with BN-apply+lrelu and residual
__global__ void edge2node_bn(const float* __restrict__ Y2, const float* __restrict__ scale,
                             const float* __restrict__ shift, const float* __restrict__ Xprev,
                             float* __restrict__ out, int E, int Vn, int C, int total) {
  int idx = blockIdx.x * blockDim.x + threadIdx.x;
  if (idx >= total) return;
  int c = idx % C;
  int v = (idx / C) % Vn;
  int n = idx / (C * Vn);
  const float* base = Y2 + ((size_t)n * E + (size_t)v * (Vn - 1)) * C + c;
  float s = 0.f;
  for (int jj = 0; jj < Vn - 1; ++jj) {
    float y = base[(size_t)jj * C] * scale[c] + shift[c];
    s += lrelu(y);
  }
  out[idx] = Xprev[idx] + s / (float)Vn;
}

// A4[n][q][a*20+t] = xns2[n][t][a*7+q] (a<18) ; A4[n][q][360+c] = xnode2[n][q][c]
__global__ void build_a4(const float* __restrict__ xns2, const float* __restrict__ xnode2,
                         float* __restrict__ A4) {
  int idx = blockIdx.x * blockDim.x + threadIdx.x;
  const int TOT = 7168 * 488;
  if (idx >= TOT) return;
  int f = idx % 488;
  int q = (idx / 488) % 7;
  int n = idx / (488 * 7);
  float v;
  if (f < 360) {
    int a = f / 20, t = f % 20;
    v = xns2[((size_t)n * 20 + t) * 126 + a * 7 + q];
  } else {
    v = xnode2[((size_t)n * 7 + q) * 128 + (f - 360)];
  }
  A4[idx] = v;
}

// ---------------------------------------------------------------------------
extern "C" void kernel_launch(void* const* d_in, const int* in_sizes, int n_in,
                              void* d_out, int out_size, void* d_ws, size_t ws_size,
                              hipStream_t stream) {
  (void)in_sizes; (void)n_in; (void)out_size; (void)ws_size;
  const float* x      = (const float*)d_in[0];
  const float* conv_w = (const float*)d_in[5];
  const float* conv_b = (const float*)d_in[6];
  const float* bn2g   = (const float*)d_in[7];
  const float* bn2b   = (const float*)d_in[8];
  // mlp params in insertion order: w1, b1, w2, b2, g, b
  const float* m1_w1 = (const float*)d_in[9],  * m1_b1 = (const float*)d_in[10];
  const float* m1_w2 = (const float*)d_in[11], * m1_b2 = (const float*)d_in[12];
  const float* m1_g  = (const float*)d_in[13], * m1_b  = (const float*)d_in[14];
  const float* m2_w1 = (const float*)d_in[15], * m2_b1 = (const float*)d_in[16];
  const float* m2_w2 = (const float*)d_in[17], * m2_b2 = (const float*)d_in[18];
  const float* m2_g  = (const float*)d_in[19], * m2_b  = (const float*)d_in[20];
  const float* s1_w1 = (const float*)d_in[21], * s1_b1 = (const float*)d_in[22];
  const float* s1_w2 = (const float*)d_in[23], * s1_b2 = (const float*)d_in[24];
  const float* s1_g  = (const float*)d_in[25], * s1_b  = (const float*)d_in[26];
  const float* s2_w1 = (const float*)d_in[27], * s2_b1 = (const float*)d_in[28];
  const float* s2_w2 = (const float*)d_in[29], * s2_b2 = (const float*)d_in[30];
  const float* s2_g  = (const float*)d_in[31], * s2_b  = (const float*)d_in[32];
  const float* m4_w1 = (const float*)d_in[33], * m4_b1 = (const float*)d_in[34];
  const float* m4_w2 = (const float*)d_in[35], * m4_b2 = (const float*)d_in[36];
  const float* m4_g  = (const float*)d_in[37], * m4_b  = (const float*)d_in[38];

  const int N = 1024, V = 7, TV2 = 20;
  const int R1 = N * V;        // 7168
  const int R1S = N * TV2;     // 20480
  const int E_J = V * (V - 1); // 42
  const int E_S = TV2 * (TV2 - 1); // 380
  const int R2 = N * E_J;      // 43008
  const int R2S = N * E_S;     // 389120
  const int HN = 128, HS = 126;

  float* Wf = (float*)d_ws;
  size_t off = 0;
  size_t oPART = off; off += 65536;
  size_t oSC   = off; off += 128;
  size_t oSH   = off; off += 128;
  size_t oSC2  = off; off += 16;
  size_t oSH2  = off; off += 16;
  size_t oH    = off; off += (size_t)N * 2240;      // h nhwc  (x1/x2 views)
  size_t oXNODE  = off; off += (size_t)R1 * HN;
  size_t oXNS    = off; off += (size_t)R1S * HS;
  size_t oXNODE2 = off; off += (size_t)R1 * HN;
  size_t oXNS2   = off; off += (size_t)R1S * HS;
  size_t oA4     = off; off += (size_t)R1 * 488;
  size_t oY1     = off; off += 2580480;             // big region, part 1
  size_t oY2     = off; off += (size_t)R2S * HS;    // big region, part 2 (196 MB)
  // h0 (conv pre-BN, 2.29M floats) lives in the (free at that time) oY2 region
  float* h0 = Wf + oY2;

  const int FLAG_ACC = 1, FLAG_LRELU = 2;
  auto gemm = [&](const float* A, int lda, const float* Wt, int ldw, const float* bias,
                  float* C, int ldc, int R, int K, int O, int flags) {
    dim3 grid(R / 32, ((O + 15) / 16 + 3) / 4);
    gemm_wmma_f32<<<grid, dim3(32, 4), 0, stream>>>(A, lda, Wt, ldw, bias, C, ldc, R, K, O, flags);
  };
  auto bnstats = [&](const float* Y, int R, int O, const float* g, const float* b) {
    colsum_partial<<<256, 128, 0, stream>>>(Y, R, O, Wf + oPART);
    colstats_final<<<1, 128, 0, stream>>>(Wf + oPART, 256, R, O, g, b, Wf + oSC, Wf + oSH);
  };

  // 1) conv + bias -> h0 (NCHW), BN2d stats, apply + transpose -> h (n,t,v,c)
  conv_time<<<(N * 2240 + 255) / 256, 256, 0, stream>>>(x, conv_w, conv_b, h0);
  bn2d_stats<<<16, 256, 0, stream>>>(h0, bn2g, bn2b, Wf + oSC2, Wf + oSH2);
  bn2d_apply_nhwc<<<(N * 2240 + 255) / 256, 256, 0, stream>>>(h0, Wf + oSC2, Wf + oSH2, Wf + oH);

  // 2) MLP1: x1 = h viewed [7168,320]
  gemm(Wf + oH, 320, m1_w1, 128, m1_b1, Wf + oY1, 128, R1, 320, 128, FLAG_LRELU);
  gemm(Wf + oY1, 128, m1_w2, 128, m1_b2, Wf + oY2, 128, R1, 128, 128, 0);
  gemm(Wf + oH, 320, m1_w2 + 128 * 128, 128, nullptr, Wf + oY2, 128, R1, 320, 128, FLAG_ACC);
  bnstats(Wf + oY2, R1, 128, m1_g, m1_b);
  bn_apply_lrelu<<<(R1 * 128 + 255) / 256, 256, 0, stream>>>(Wf + oY2, Wf + oSC, Wf + oSH,
                                                             Wf + oXNODE, 128, R1 * 128);

  // 3) MLP1s: x2 = h viewed [20480,112]
  gemm(Wf + oH, 112, s1_w1, 126, s1_b1, Wf + oY1, 126, R1S, 112, 126, FLAG_LRELU);
  gemm(Wf + oY1, 126, s1_w2, 126, s1_b2, Wf + oY2, 126, R1S, 126, 126, 0);
  gemm(Wf + oH, 112, s1_w2 + 126 * 126, 126, nullptr, Wf + oY2, 126, R1S, 112, 126, FLAG_ACC);
  bnstats(Wf + oY2, R1S, 126, s1_g, s1_b);
  bn_apply_lrelu<<<(R1S * 126 + 255) / 256, 256, 0, stream>>>(Wf + oY2, Wf + oSC, Wf + oSH,
                                                              Wf + oXNS, 126, R1S * 126);

  // 4) joint edges: fused node2edge + MLP2 -> Y2 (pre-BN), BN stats, edge2node+residual
  edge_mlp_fused<<<R2 / 32, dim3(32, 8), (32 * 256 + 32 * 128) * 4, stream>>>(
      Wf + oXNODE, m2_w1, m2_b1, m2_w2, m2_b2, Wf + oY2, E_J, V, 128, 128);
  bnstats(Wf + oY2, R2, 128, m2_g, m2_b);
  edge2node_bn<<<(R1 * 128 + 255) / 256, 256, 0, stream>>>(
      Wf + oY2, Wf + oSC, Wf + oSH, Wf + oXNODE, Wf + oXNODE2, E_J, V, 128, R1 * 128);

  // 5) speed edges: fused node2edge + MLP2s (the 66-GFLOP hot spot)
  edge_mlp_fused<<<R2S / 32, dim3(32, 8), (32 * 252 + 32 * 128) * 4, stream>>>(
      Wf + oXNS, s2_w1, s2_b1, s2_w2, s2_b2, Wf + oY2, E_S, TV2, 126, 126);
  bnstats(Wf + oY2, R2S, 126, s2_g, s2_b);
  edge2node_bn<<<(R1S * 126 + 255) / 256, 256, 0, stream>>>(
      Wf + oY2, Wf + oSC, Wf + oSH, Wf + oXNS, Wf + oXNS2, E_S, TV2, 126, R1S * 126);

  // 6) transpose/concat -> A4 [7168,488]; final MLP4s -> d_out
  build_a4<<<(R1 * 488 + 255) / 256, 256, 0, stream>>>(Wf + oXNS2, Wf + oXNODE2, Wf + oA4);
  gemm(Wf + oA4, 488, m4_w1, 128, m4_b1, Wf + oY1, 128, R1, 488, 128, FLAG_LRELU);
  gemm(Wf + oY1, 128, m4_w2, 128, m4_b2, Wf + oY2, 128, R1, 128, 128, 0);
  gemm(Wf + oA4, 488, m4_w2 + 128 * 128, 128, nullptr, Wf + oY2, 128, R1, 488, 128, FLAG_ACC);
  bnstats(Wf + oY2, R1, 128, m4_g, m4_b);
  bn_apply_lrelu<<<(R1 * 128 + 255) / 256, 256, 0, stream>>>(Wf + oY2, Wf + oSC, Wf + oSH,
                                                             (float*)d_out, 128, R1 * 128);
}